// Merge_MixtralSparseMoeBlock_9637906612731
// MI455X (gfx1250) — compile-verified
//
#include <hip/hip_runtime.h>
#include <math.h>
#include <stdint.h>

typedef __attribute__((ext_vector_type(2))) float v2f;
typedef __attribute__((ext_vector_type(8))) float v8f;

#define TDIM 2048
#define HDIM 2048
#define IDIM 7168
#define EXP  8
#define RDIM 159
#define RPAD 160

#define BM   128  // block tile rows (MSUB=2): weight re-read factor T/BM = 16x
#define BK   32
#define LSTR 36   // LDS row stride (floats): float4-aligned, 16 consecutive rows -> 16 distinct banks

// ---------------------------------------------------------------------------
// Async global->LDS staging (CDNA5 ASYNC path): 16 bytes per lane,
// no VGPR round-trip, tracked by ASYNCcnt.
// ---------------------------------------------------------------------------
__device__ __forceinline__ void async_ld16(float* lds, const float* gsrc) {
    asm volatile("global_load_async_to_lds_b128 %0, %1, off"
                 :: "v"((unsigned)(uintptr_t)lds),
                    "v"((unsigned long long)(uintptr_t)gsrc)
                 : "memory");
}
__device__ __forceinline__ void wait_async0() {
    asm volatile("s_wait_asynccnt 0x0" ::: "memory");
}

// Stage nPass*32 rows x 32 cols, fully regular case (no guards, aligned).
__device__ __forceinline__ void stage_async(const float* __restrict__ S, int ld,
    int gbase, int k0, float* dst, int nPass, int ldRow, int ldK)
{
    #pragma unroll
    for (int i = 0; i < nPass; i++) {
        const int rl = ldRow + 32 * i;
        async_ld16(&dst[rl * LSTR + ldK], S + (size_t)(gbase + rl) * ld + k0 + ldK);
    }
}

// Guarded scalar staging for ragged K / unaligned ldb / partial N tiles.
__device__ __forceinline__ void stage_sync(const float* __restrict__ S, int ld,
    int gbase, int gtotal, int k0, int kRem, float* dst, int nPass, int ldRow, int ldK)
{
    #pragma unroll
    for (int i = 0; i < nPass; i++) {
        const int rl = ldRow + 32 * i;
        const int g  = gbase + rl;
        float4 v = make_float4(0.f, 0.f, 0.f, 0.f);
        if (g < gtotal) {
            const float* s = S + (size_t)g * ld + k0 + ldK;
            if (kRem >= BK && ((ld & 3) == 0)) {
                v = *(const float4*)s;
            } else {
                v.x = (ldK + 0 < kRem) ? s[0] : 0.f;
                v.y = (ldK + 1 < kRem) ? s[1] : 0.f;
                v.z = (ldK + 2 < kRem) ? s[2] : 0.f;
                v.w = (ldK + 3 < kRem) ? s[3] : 0.f;
            }
        }
        *(float4*)&dst[rl * LSTR + ldK] = v;
    }
}

#define WAVE_IDX()                              \
    const int tid   = threadIdx.x;              \
    const int lane  = tid & 31;                 \
    const int wv    = tid >> 5;                 \
    const int waveM = wv >> 1;                  \
    const int waveN = wv & 1;                   \
    const int lr    = lane & 15;                \
    const int hi    = lane >> 4;                \
    const int ldRow = tid >> 3;                 \
    const int ldK   = (tid & 7) * 4;            \
    (void)waveM; (void)waveN; (void)lr; (void)hi;

// ---------------------------------------------------------------------------
// Generic: acc[m][s] += A[128 rows] @ Bt[NSUB*32 cols]^T.
// 8 waves in 4(M)x2(N); each wave owns (MSUB*16) x (NSUB*16)
// -> MSUB*NSUB independent WMMA chains (8 for <2,4>).
// ---------------------------------------------------------------------------
template<int MSUB, int NSUB>
__device__ __forceinline__ void gemm_block_accum(
    const float* __restrict__ A, int lda,
    const float* __restrict__ Bt, int ldb, int Ntotal,
    int K, int rowBase, int colBase,
    float* As, float* Bs, v8f (*acc)[NSUB])
{
    WAVE_IDX();
    const bool asyncOK = ((K & (BK - 1)) == 0) && ((ldb & 3) == 0) &&
                         (colBase + NSUB * 32 <= Ntotal);

    for (int k0 = 0; k0 < K; k0 += BK) {
        const int kRem = K - k0;
        if (asyncOK) {
            stage_async(A,  lda, rowBase, k0, As, 2 * MSUB, ldRow, ldK);
            stage_async(Bt, ldb, colBase, k0, Bs, NSUB,     ldRow, ldK);
            wait_async0();
        } else {
            stage_sync(A,  lda, rowBase, TDIM,   k0, kRem, As, 2 * MSUB, ldRow, ldK);
            stage_sync(Bt, ldb, colBase, Ntotal, k0, kRem, Bs, NSUB,     ldRow, ldK);
        }
        __syncthreads();

        const float* ap[MSUB];
        const float* bp[NSUB];
        #pragma unroll
        for (int m = 0; m < MSUB; m++)
            ap[m] = &As[(waveM * (MSUB * 16) + m * 16 + lr) * LSTR + hi * 2];
        #pragma unroll
        for (int s = 0; s < NSUB; s++)
            bp[s] = &Bs[((waveN * NSUB + s) * 16 + lr) * LSTR + hi * 2];

        #pragma unroll
        for (int kk = 0; kk < BK; kk += 4) {
            v2f a[MSUB], b[NSUB];
            #pragma unroll
            for (int m = 0; m < MSUB; m++) { a[m].x = ap[m][kk]; a[m].y = ap[m][kk + 1]; }
            #pragma unroll
            for (int s = 0; s < NSUB; s++) { b[s].x = bp[s][kk]; b[s].y = bp[s][kk + 1]; }
            #pragma unroll
            for (int m = 0; m < MSUB; m++)
                #pragma unroll
                for (int s = 0; s < NSUB; s++)
                    acc[m][s] = __builtin_amdgcn_wmma_f32_16x16x4_f32(
                        false, a[m], false, b[s], (short)0, acc[m][s], false, false);
        }
        __syncthreads();
    }
}

// ---------------------------------------------------------------------------
// Dual weight streams sharing one staged A tile (mid kernel main phase):
// acc1 += A@B1^T, acc2 += A@B2^T  -> 8 independent WMMA chains per wave.
// ---------------------------------------------------------------------------
__device__ __forceinline__ void gemm_block_accum_dual(
    const float* __restrict__ A, int lda,
    const float* __restrict__ B1, const float* __restrict__ B2,
    int ldb, int Ntotal, int K, int rowBase, int colBase,
    float* As, float* Bs1, float* Bs2,
    v8f (*acc1)[2], v8f (*acc2)[2])
{
    WAVE_IDX();
    const bool asyncOK = ((K & (BK - 1)) == 0) && ((ldb & 3) == 0) &&
                         (colBase + 64 <= Ntotal);

    for (int k0 = 0; k0 < K; k0 += BK) {
        const int kRem = K - k0;
        if (asyncOK) {
            stage_async(A,  lda, rowBase, k0, As,  4, ldRow, ldK);
            stage_async(B1, ldb, colBase, k0, Bs1, 2, ldRow, ldK);
            stage_async(B2, ldb, colBase, k0, Bs2, 2, ldRow, ldK);
            wait_async0();
        } else {
            stage_sync(A,  lda, rowBase, TDIM,   k0, kRem, As,  4, ldRow, ldK);
            stage_sync(B1, ldb, colBase, Ntotal, k0, kRem, Bs1, 2, ldRow, ldK);
            stage_sync(B2, ldb, colBase, Ntotal, k0, kRem, Bs2, 2, ldRow, ldK);
        }
        __syncthreads();

        const float* ap[2];
        const float* b1p[2];
        const float* b2p[2];
        #pragma unroll
        for (int m = 0; m < 2; m++)
            ap[m] = &As[(waveM * 32 + m * 16 + lr) * LSTR + hi * 2];
        #pragma unroll
        for (int s = 0; s < 2; s++) {
            b1p[s] = &Bs1[((waveN * 2 + s) * 16 + lr) * LSTR + hi * 2];
            b2p[s] = &Bs2[((waveN * 2 + s) * 16 + lr) * LSTR + hi * 2];
        }

        #pragma unroll
        for (int kk = 0; kk < BK; kk += 4) {
            v2f a[2], b1[2], b2[2];
            #pragma unroll
            for (int m = 0; m < 2; m++) { a[m].x  = ap[m][kk];  a[m].y  = ap[m][kk + 1]; }
            #pragma unroll
            for (int s = 0; s < 2; s++) { b1[s].x = b1p[s][kk]; b1[s].y = b1p[s][kk + 1];
                                          b2[s].x = b2p[s][kk]; b2[s].y = b2p[s][kk + 1]; }
            #pragma unroll
            for (int m = 0; m < 2; m++)
                #pragma unroll
                for (int s = 0; s < 2; s++) {
                    acc1[m][s] = __builtin_amdgcn_wmma_f32_16x16x4_f32(
                        false, a[m], false, b1[s], (short)0, acc1[m][s], false, false);
                    acc2[m][s] = __builtin_amdgcn_wmma_f32_16x16x4_f32(
                        false, a[m], false, b2[s], (short)0, acc2[m][s], false, false);
                }
        }
        __syncthreads();
    }
}

// ---------------------------------------------------------------------------
// Router: logits = x @ gate_w^T ; softmax ; top-2 ; renorm ; dense comb[T,E]
// ---------------------------------------------------------------------------
__global__ void __launch_bounds__(256) router_kernel(
    const float* __restrict__ x, const float* __restrict__ gw,
    float* __restrict__ logits, float* __restrict__ comb)
{
    const int lane = threadIdx.x & 31;
    const int t    = blockIdx.x * 8 + (threadIdx.x >> 5);

    float a[EXP];
    #pragma unroll
    for (int e = 0; e < EXP; e++) a[e] = 0.f;
    for (int h = lane; h < HDIM; h += 32) {
        const float xv = x[(size_t)t * HDIM + h];
        #pragma unroll
        for (int e = 0; e < EXP; e++) a[e] += xv * gw[(size_t)e * HDIM + h];
    }
    #pragma unroll
    for (int off = 16; off > 0; off >>= 1) {
        #pragma unroll
        for (int e = 0; e < EXP; e++) a[e] += __shfl_xor(a[e], off, 32);
    }
    if (lane == 0) {
        float m = a[0];
        #pragma unroll
        for (int e = 1; e < EXP; e++) m = fmaxf(m, a[e]);
        float p[EXP], s = 0.f;
        #pragma unroll
        for (int e = 0; e < EXP; e++) { p[e] = expf(a[e] - m); s += p[e]; }
        #pragma unroll
        for (int e = 0; e < EXP; e++) p[e] /= s;
        int i1 = 0;
        #pragma unroll
        for (int e = 1; e < EXP; e++) if (p[e] > p[i1]) i1 = e;
        int i2 = (i1 == 0) ? 1 : 0;
        #pragma unroll
        for (int e = 0; e < EXP; e++) if (e != i1 && p[e] > p[i2]) i2 = e;
        const float ws = p[i1] + p[i2];
        #pragma unroll
        for (int e = 0; e < EXP; e++) {
            comb[(size_t)t * EXP + e]   = (e == i1) ? p[i1] / ws : ((e == i2) ? p[i2] / ws : 0.f);
            logits[(size_t)t * EXP + e] = a[e];
        }
    }
}

// xv{1,3} = x @ V{1,3}^T : [T, R] into padded [T, RPAD].  grid.z selects V1/V3.
__global__ void __launch_bounds__(256) xv_kernel(
    const float* __restrict__ x,
    const float* __restrict__ V1e, const float* __restrict__ V3e,
    float* __restrict__ xv1, float* __restrict__ xv3)
{
    __shared__ float As[BM * LSTR];
    __shared__ float Bs[64 * LSTR];
    const int rowBase = blockIdx.x * BM;
    const int colBase = blockIdx.y * 64;
    const float* V   = (blockIdx.z == 0) ? V1e : V3e;
    float*       dst = (blockIdx.z == 0) ? xv1 : xv3;

    v8f z = {0,0,0,0,0,0,0,0};
    v8f acc[2][2] = {{z, z}, {z, z}};
    gemm_block_accum<2, 2>(x, HDIM, V, HDIM, RDIM, HDIM, rowBase, colBase, As, Bs, acc);

    WAVE_IDX();
    #pragma unroll
    for (int m = 0; m < 2; m++) {
        const int rbase = rowBase + waveM * 32 + m * 16 + hi * 8;
        #pragma unroll
        for (int s = 0; s < 2; s++) {
            const int c = colBase + waveN * 32 + s * 16 + lr;
            if (c < RDIM) {
                #pragma unroll
                for (int j = 0; j < 8; j++)
                    dst[(size_t)(rbase + j) * RPAD + c] = acc[m][s][j];
            }
        }
    }
}

// hmid = silu(x@W1^T + xv1@U1^T) * (x@W3^T + xv3@U3^T)  : [T, I]
__global__ void __launch_bounds__(256) mid_kernel(
    const float* __restrict__ x,
    const float* __restrict__ xv1, const float* __restrict__ xv3,
    const float* __restrict__ W1e, const float* __restrict__ U1e,
    const float* __restrict__ W3e, const float* __restrict__ U3e,
    float* __restrict__ hmid)
{
    __shared__ float As[BM * LSTR];
    __shared__ float Bs1[64 * LSTR];
    __shared__ float Bs2[64 * LSTR];
    const int rowBase = blockIdx.x * BM;
    const int colBase = blockIdx.y * 64;

    v8f z = {0,0,0,0,0,0,0,0};
    v8f accg[2][2] = {{z, z}, {z, z}};
    v8f accu[2][2] = {{z, z}, {z, z}};
    // W1/W3 share the x tile: one staged A, two async weight streams, 8 WMMA chains.
    gemm_block_accum_dual(x, HDIM, W1e, W3e, HDIM, IDIM, HDIM, rowBase, colBase,
                          As, Bs1, Bs2, accg, accu);
    // low-rank deltas (ragged K = 159, unaligned ldb -> guarded sync staging; tiny)
    gemm_block_accum<2, 2>(xv1, RPAD, U1e, RDIM, IDIM, RDIM, rowBase, colBase, As, Bs1, accg);
    gemm_block_accum<2, 2>(xv3, RPAD, U3e, RDIM, IDIM, RDIM, rowBase, colBase, As, Bs1, accu);

    WAVE_IDX();
    #pragma unroll
    for (int m = 0; m < 2; m++) {
        const int rbase = rowBase + waveM * 32 + m * 16 + hi * 8;
        #pragma unroll
        for (int s = 0; s < 2; s++) {
            const int c = colBase + waveN * 32 + s * 16 + lr;
            #pragma unroll
            for (int j = 0; j < 8; j++) {
                const float g = accg[m][s][j], u = accu[m][s][j];
                hmid[(size_t)(rbase + j) * IDIM + c] = (g / (1.f + expf(-g))) * u;
            }
        }
    }
}

// hv2 = hmid @ V2^T : [T, R] padded to [T, RPAD]
__global__ void __launch_bounds__(256) hv2_kernel(
    const float* __restrict__ hmid, const float* __restrict__ V2e, float* __restrict__ hv2)
{
    __shared__ float As[BM * LSTR];
    __shared__ float Bs[64 * LSTR];
    const int rowBase = blockIdx.x * BM;
    const int colBase = blockIdx.y * 64;

    v8f z = {0,0,0,0,0,0,0,0};
    v8f acc[2][2] = {{z, z}, {z, z}};
    gemm_block_accum<2, 2>(hmid, IDIM, V2e, IDIM, RDIM, IDIM, rowBase, colBase, As, Bs, acc);

    WAVE_IDX();
    #pragma unroll
    for (int m = 0; m < 2; m++) {
        const int rbase = rowBase + waveM * 32 + m * 16 + hi * 8;
        #pragma unroll
        for (int s = 0; s < 2; s++) {
            const int c = colBase + waveN * 32 + s * 16 + lr;
            if (c < RDIM) {
                #pragma unroll
                for (int j = 0; j < 8; j++)
                    hv2[(size_t)(rbase + j) * RPAD + c] = acc[m][s][j];
            }
        }
    }
}

// out (+)= comb[:,e] * (hmid@W2^T + hv2@U2^T)   128x128 tile, 8 WMMA chains/wave
__global__ void __launch_bounds__(256) out_kernel(
    const float* __restrict__ hmid, const float* __restrict__ hv2,
    const float* __restrict__ W2e, const float* __restrict__ U2e,
    const float* __restrict__ comb, int e, int first, float* __restrict__ out)
{
    __shared__ float As[BM * LSTR];
    __shared__ float Bs[128 * LSTR];
    const int rowBase = blockIdx.x * BM;
    const int colBase = blockIdx.y * 128;

    v8f z = {0,0,0,0,0,0,0,0};
    v8f acc[2][4] = {{z, z, z, z}, {z, z, z, z}};
    gemm_block_accum<2, 4>(hmid, IDIM, W2e, IDIM, HDIM, IDIM, rowBase, colBase, As, Bs, acc);
    gemm_block_accum<2, 4>(hv2,  RPAD, U2e, RDIM, HDIM, RDIM, rowBase, colBase, As, Bs, acc);

    WAVE_IDX();
    #pragma unroll
    for (int m = 0; m < 2; m++) {
        const int rbase = rowBase + waveM * 32 + m * 16 + hi * 8;
        #pragma unroll
        for (int s = 0; s < 4; s++) {
            const int c = colBase + waveN * 64 + s * 16 + lr;
            #pragma unroll
            for (int j = 0; j < 8; j++) {
                const int r = rbase + j;
                const float w = comb[(size_t)r * EXP + e];
                const size_t o = (size_t)r * HDIM + c;
                if (first) out[o]  = w * acc[m][s][j];
                else       out[o] += w * acc[m][s][j];
            }
        }
    }
}

extern "C" void kernel_launch(void* const* d_in, const int* in_sizes, int n_in,
                              void* d_out, int out_size, void* d_ws, size_t ws_size,
                              hipStream_t stream) {
    (void)in_sizes; (void)n_in; (void)out_size; (void)ws_size;
    const float* x      = (const float*)d_in[0];   // [T, H]
    const float* gate_w = (const float*)d_in[1];   // [E, H]
    const float* W1     = (const float*)d_in[2];   // [E, I, H]
    const float* W2     = (const float*)d_in[3];   // [E, H, I]
    const float* W3     = (const float*)d_in[4];   // [E, I, H]
    const float* U1     = (const float*)d_in[5];   // [E, I, R]
    const float* V1     = (const float*)d_in[6];   // [E, R, H]
    const float* U2     = (const float*)d_in[7];   // [E, H, R]
    const float* V2     = (const float*)d_in[8];   // [E, R, I]
    const float* U3     = (const float*)d_in[9];   // [E, I, R]
    const float* V3     = (const float*)d_in[10];  // [E, R, H]

    float* out    = (float*)d_out;                       // [T, H]
    float* logits = out + (size_t)TDIM * HDIM;           // [T, E]

    float* comb = (float*)d_ws;                          // [T, E]
    float* xv1  = comb + (size_t)TDIM * EXP;             // [T, RPAD]
    float* xv3  = xv1  + (size_t)TDIM * RPAD;            // [T, RPAD]
    float* hv2  = xv3  + (size_t)TDIM * RPAD;            // [T, RPAD]
    float* hmid = hv2  + (size_t)TDIM * RPAD;            // [T, I]

    router_kernel<<<TDIM / 8, 256, 0, stream>>>(x, gate_w, logits, comb);

    const dim3 blk(256);
    const dim3 gXV(TDIM / BM, (RDIM + 63) / 64, 2);      // (16, 3, 2)
    const dim3 gMID(TDIM / BM, IDIM / 64);               // (16, 112)
    const dim3 gHV(TDIM / BM, (RDIM + 63) / 64);         // (16, 3)
    const dim3 gOUT(TDIM / BM, HDIM / 128);              // (16, 16)

    for (int e = 0; e < EXP; e++) {
        const float* W1e = W1 + (size_t)e * IDIM * HDIM;
        const float* W2e = W2 + (size_t)e * HDIM * IDIM;
        const float* W3e = W3 + (size_t)e * IDIM * HDIM;
        const float* U1e = U1 + (size_t)e * IDIM * RDIM;
        const float* V1e = V1 + (size_t)e * RDIM * HDIM;
        const float* U2e = U2 + (size_t)e * HDIM * RDIM;
        const float* V2e = V2 + (size_t)e * RDIM * IDIM;
        const float* U3e = U3 + (size_t)e * IDIM * RDIM;
        const float* V3e = V3 + (size_t)e * RDIM * HDIM;

        xv_kernel <<<gXV,  blk, 0, stream>>>(x, V1e, V3e, xv1, xv3);
        mid_kernel<<<gMID, blk, 0, stream>>>(x, xv1, xv3, W1e, U1e, W3e, U3e, hmid);
        hv2_kernel<<<gHV,  blk, 0, stream>>>(hmid, V2e, hv2);
        out_kernel<<<gOUT, blk, 0, stream>>>(hmid, hv2, W2e, U2e, comb, e, (e == 0) ? 1 : 0, out);
    }
}